// LSSLKernel_84361747628478
// MI455X (gfx1250) — compile-verified
//
#include <hip/hip_runtime.h>
#include <hip/hip_bf16.h>

typedef __attribute__((ext_vector_type(2))) float v2f;
typedef __attribute__((ext_vector_type(8))) float v8f;

struct C2 { float re, im; };
__device__ __forceinline__ C2 cmul(C2 a, C2 b) {
    return { a.re * b.re - a.im * b.im, a.re * b.im + a.im * b.re };
}
__device__ __forceinline__ C2 cpow(C2 z, int e) {
    C2 r{1.0f, 0.0f};
    C2 b = z;
    while (e) {
        if (e & 1) r = cmul(r, b);
        b = cmul(b, b);
        e >>= 1;
    }
    return r;
}

#define DM 512
#define SD 64
#define NSLICE (SD / 4)   // 16 K-slices of depth 4 for V_WMMA_F32_16X16X4_F32

__global__ __launch_bounds__(32)
void lssl_vandermonde_wmma(const float* __restrict__ Lre, const float* __restrict__ Lim,
                           const float* __restrict__ Bre, const float* __restrict__ Bim,
                           const float* __restrict__ Cre, const float* __restrict__ Cim,
                           const float* __restrict__ Din, const float* __restrict__ log_dt,
                           float* __restrict__ out, int L, int n_tiles) {
    // Per-state discretized params (shared across the wave)
    __shared__ float aR[SD], aI[SD], wR[SD], wI[SD];
    __shared__ float a16R[SD], a16I[SD];     // a^16
    __shared__ float s16R[SD], s16I[SD];     // a^256 (tile-advance step)
    // P[l0][n] = w[n]*a[n]^l0 : real part, and NEGATED imag part (f32 WMMA has no A-neg)
    __shared__ float Pre[16 * SD], Pni[16 * SD];
    // Q[n][col] = (a[n]^16)^(16*t + col) for the current tile t
    __shared__ float Qre[SD * 16], Qim[SD * 16];

    const int d    = blockIdx.x;
    const int lane = threadIdx.x;

    // ---- Phase A: bilinear (Tustin) discretization, 2 states per lane ----
    const float dt = expf(log_dt[d]);
    for (int n = lane; n < SD; n += 32) {
        float x  = Lre[d * SD + n];
        float sp = fmaxf(x, 0.0f) + log1pf(expf(-fabsf(x)));   // stable softplus
        C2 lam  { -sp, Lim[d * SD + n] };
        C2 half { 0.5f * dt * lam.re, 0.5f * dt * lam.im };
        C2 den  { 1.0f - half.re, -half.im };
        float inv = 1.0f / (den.re * den.re + den.im * den.im);
        C2 num  { 1.0f + half.re, half.im };
        C2 a { (num.re * den.re + num.im * den.im) * inv,
               (num.im * den.re - num.re * den.im) * inv };     // (1+h)/(1-h)
        C2 dtden { dt * den.re * inv, -dt * den.im * inv };     // dt/den
        C2 Bv { Bre[d * SD + n], Bim[d * SD + n] };
        C2 bb = cmul(dtden, Bv);
        C2 Cv { Cre[d * SD + n], Cim[d * SD + n] };
        C2 w  = cmul(Cv, bb);
        aR[n] = a.re;  aI[n] = a.im;  wR[n] = w.re;  wI[n] = w.im;
        C2 a16 = a;
        #pragma unroll
        for (int i = 0; i < 4; ++i) a16 = cmul(a16, a16);       // a^16
        a16R[n] = a16.re; a16I[n] = a16.im;
        C2 s16 = a16;
        #pragma unroll
        for (int i = 0; i < 4; ++i) s16 = cmul(s16, s16);       // a^256
        s16R[n] = s16.re; s16I[n] = s16.im;
    }
    __syncthreads();

    // ---- Phase B: P[l0][n] = w[n] * a[n]^l0 (idx == l0*64 + n) ----
    for (int idx = lane; idx < 16 * SD; idx += 32) {
        int l0 = idx >> 6;
        int n  = idx & (SD - 1);
        C2 a { aR[n], aI[n] };
        C2 w { wR[n], wI[n] };
        C2 pw = cmul(w, cpow(a, l0));
        Pre[idx] = pw.re;
        Pni[idx] = -pw.im;
    }
    // ---- Phase C: Q[n][col] = (a[n]^16)^col  (idx == n*16 + col) ----
    for (int idx = lane; idx < SD * 16; idx += 32) {
        int n   = idx >> 4;
        int col = idx & 15;
        C2 a16 { a16R[n], a16I[n] };
        C2 q = cpow(a16, col);
        Qre[idx] = q.re;
        Qim[idx] = q.im;
    }
    __syncthreads();

    // ---- A fragments (held in registers for all tiles) ----
    // f32 16x4 A layout: lanes 0-15 -> M=lane, v0=K0, v1=K1; lanes 16-31 -> M=lane-16, v0=K2, v1=K3
    const int m   = lane & 15;
    const int off = (lane < 16) ? 0 : 2;
    v2f aRe[NSLICE], aNi[NSLICE];
    #pragma unroll
    for (int kk = 0; kk < NSLICE; ++kk) {
        int n0 = 4 * kk + off;
        aRe[kk] = (v2f){ Pre[m * SD + n0], Pre[m * SD + n0 + 1] };
        aNi[kk] = (v2f){ Pni[m * SD + n0], Pni[m * SD + n0 + 1] };
    }

    // ---- Tile loop over groups of 16 l1 columns ----
    const int col = lane & 15;
    for (int t = 0; t < n_tiles; ++t) {
        v8f acc = {};
        #pragma unroll
        for (int kk = 0; kk < NSLICE; ++kk) {
            int n0 = 4 * kk + off;
            // f32 4x16 B layout: lanes 0-15 -> N=lane, rows K0/K1; lanes 16-31 -> N=lane-16, rows K2/K3
            v2f bRe = (v2f){ Qre[n0 * 16 + col], Qre[(n0 + 1) * 16 + col] };
            v2f bIm = (v2f){ Qim[n0 * 16 + col], Qim[(n0 + 1) * 16 + col] };
            acc = __builtin_amdgcn_wmma_f32_16x16x4_f32(false, aRe[kk], false, bRe,
                                                        (short)0, acc, false, false);
            acc = __builtin_amdgcn_wmma_f32_16x16x4_f32(false, aNi[kk], false, bIm,
                                                        (short)0, acc, false, false);
        }
        // Store: C/D layout -> lanes 0-15: N=lane, VGPR r = row M=r; lanes 16-31: M=r+8
        const int mbase = (lane < 16) ? 0 : 8;
        const int l1    = 16 * t + col;
        #pragma unroll
        for (int r = 0; r < 8; ++r) {
            int l0 = mbase + r;
            out[(size_t)d * L + (l0 + 16 * l1)] = acc[r];
        }
        __syncthreads();
        // Advance Q one tile: Q *= a^256
        if (t + 1 < n_tiles) {
            for (int idx = lane; idx < SD * 16; idx += 32) {
                int n = idx >> 4;
                C2 q { Qre[idx], Qim[idx] };
                C2 s { s16R[n], s16I[n] };
                C2 nq = cmul(q, s);
                Qre[idx] = nq.re;
                Qim[idx] = nq.im;
            }
        }
        __syncthreads();
    }

    // ---- Passthrough D output, concatenated after K ----
    if (lane == 0) out[(size_t)gridDim.x * L + d] = Din[d];
}

extern "C" void kernel_launch(void* const* d_in, const int* in_sizes, int n_in,
                              void* d_out, int out_size, void* d_ws, size_t ws_size,
                              hipStream_t stream) {
    const float* Lre    = (const float*)d_in[0];
    const float* Lim    = (const float*)d_in[1];
    const float* Bre    = (const float*)d_in[2];
    const float* Bim    = (const float*)d_in[3];
    const float* Cre    = (const float*)d_in[4];
    const float* Cim    = (const float*)d_in[5];
    const float* Din    = (const float*)d_in[6];
    const float* log_dt = (const float*)d_in[7];
    float* out = (float*)d_out;

    const int d_model = in_sizes[6];                 // |D| == d_model (512)
    const int L       = out_size / d_model - 1;      // out = d_model*L + d_model
    const int n_tiles = L / 256;                     // 16 l0-rows x 16 cols per tile

    lssl_vandermonde_wmma<<<dim3(d_model), dim3(32), 0, stream>>>(
        Lre, Lim, Bre, Bim, Cre, Cim, Din, log_dt, out, L, n_tiles);
}